// VNL_Loss_86285892977289
// MI455X (gfx1250) — compile-verified
//
#include <hip/hip_runtime.h>
#include <math.h>

// ---------------------------------------------------------------------------
// VNL loss for MI455X (gfx1250, wave32).
// Pipeline (all device-side, graph-capturable, static launch sequence):
//   initK -> planarK -> nonplanarK -> prepK -> 4x(histK -> scanK) -> finalK
// Trimmed mean via exact 4-pass radix select on order-preserving float keys.
// Wave sums use v_wmma_f32_16x16x4_f32 (B = ones => row sums), histogram
// merge uses ds_add_f32 / global_atomic_add_f32.
// ---------------------------------------------------------------------------

namespace {
constexpr int Wd   = 1920;
constexpr int Ht   = 1080;
constexpr int NP   = 8;
constexpr int G    = 62208;          // int(108*1920*0.3)
constexpr int GN   = 124416;         // int(216*1920*0.3)
constexpr int TOT  = NP * G;         // 497664
constexpr int NG   = NP + 1;         // 9 groups (8 planar + 1 nonplanar)
constexpr int NELEM = TOT + GN;      // 622080 (divisible by 256)
}

typedef __attribute__((ext_vector_type(2))) float v2f;
typedef __attribute__((ext_vector_type(8))) float v8f;

struct GroupState {
  unsigned int V;        // valid count
  float        S;        // sum of valid (non-NaN) losses
  unsigned int prefix;   // resolved high bytes of k-th smallest key
  unsigned int r;        // remaining rank within matching prefix
  unsigned int done;     // 1 if k==0 (nothing to drop)
  float        sum_less; // sum of values strictly below resolved prefix
  unsigned int pad0, pad1;
};

struct F3 { float x, y, z; };

__device__ __forceinline__ F3 f3sub(F3 a, F3 b){ return {a.x-b.x, a.y-b.y, a.z-b.z}; }
__device__ __forceinline__ float f3dot(F3 a, F3 b){ return a.x*b.x + a.y*b.y + a.z*b.z; }
__device__ __forceinline__ F3 f3cross(F3 a, F3 b){
  return {a.y*b.z - a.z*b.y, a.z*b.x - a.x*b.z, a.x*b.y - a.y*b.x};
}

__device__ __forceinline__ F3 unproject(const float* __restrict__ depth, int i,
                                        float fx, float fy){
  int v = i / Wd;
  int u = i - v * Wd;
  float d  = depth[i];
  float ad = fabsf(d);
  F3 p;
  p.x = (float)(u - (Wd / 2)) * ad / fx;   // ((u-960)*|d|)/fx, same assoc as ref
  p.y = (float)(v - (Ht / 2)) * ad / fy;
  p.z = d;
  return p;
}

// Exact wave32 sum via V_WMMA_F32_16X16X4_F32.
// A layout (16x4 f32): lanes 0-15: v0->A[m,0], v1->A[m,1]; lanes 16-31: v0->A[m,2], v1->A[m,3].
// With A[m,0]=x[m], A[m,2]=x[m+16], others 0, and B = all ones:
//   D[m,n] = x[m] + x[m+16] for every n.
// D layout: vgpr r, lanes 0-15 -> M=r; lanes 16-31 -> M=r+8. Fold 8 regs, xor-16 shuffle.
// EXEC must be all ones at the call site (callers guarantee full blocks, no guards).
__device__ __forceinline__ float waveSumWMMA(float x){
  v2f a; a[0] = x;     a[1] = 0.0f;
  v2f b; b[0] = 1.0f;  b[1] = 1.0f;
  v8f c = {};
  v8f d = __builtin_amdgcn_wmma_f32_16x16x4_f32(false, a, false, b,
                                                (short)0, c, false, false);
  float t = ((d[0] + d[1]) + (d[2] + d[3])) + ((d[4] + d[5]) + (d[6] + d[7]));
  t += __shfl_xor(t, 16, 32);
  return t;  // full wave sum in every lane
}

__device__ __forceinline__ bool filterMask(F3 P1, F3 P2, F3 P3, float dd){
  F3 d12 = f3sub(P2, P1), d13 = f3sub(P3, P1), d23 = f3sub(P3, P2);
  float n1s = f3dot(d12, d12), n2s = f3dot(d13, d13), n3s = f3dot(d23, d23);
  float q1 = sqrtf(n1s), q2 = sqrtf(n2s), q3 = sqrtf(n3s);
  float e12 = f3dot(d12, d13), e13 = f3dot(d12, d23), e23 = f3dot(d13, d23);
  int cnt = 0;  // count over the full symmetric 3x3 |ne| > 0.985 (diag included)
  cnt += (fabsf(n1s / (q1 * q1 + 1e-8f)) > 0.985f);
  cnt += (fabsf(n2s / (q2 * q2 + 1e-8f)) > 0.985f);
  cnt += (fabsf(n3s / (q3 * q3 + 1e-8f)) > 0.985f);
  cnt += 2 * (fabsf(e12 / (q1 * q2 + 1e-8f)) > 0.985f);
  cnt += 2 * (fabsf(e13 / (q1 * q3 + 1e-8f)) > 0.985f);
  cnt += 2 * (fabsf(e23 / (q2 * q3 + 1e-8f)) > 0.985f);
  bool mask_cos = cnt > 3;
  bool mask_pad = (P1.z > 1e-4f) && (P2.z > 1e-4f) && (P3.z > 1e-4f);
  bool mx = (fabsf(d12.x) < dd) || (fabsf(d13.x) < dd) || (fabsf(d23.x) < dd);
  bool my = (fabsf(d12.y) < dd) || (fabsf(d13.y) < dd) || (fabsf(d23.y) < dd);
  bool mz = (fabsf(d12.z) < dd) || (fabsf(d13.z) < dd) || (fabsf(d23.z) < dd);
  return mask_pad && !((mx && my && mz) || mask_cos);
}

__device__ __forceinline__ F3 normalOf(F3 P1, F3 P2, F3 P3){
  F3 n = f3cross(f3sub(P2, P1), f3sub(P3, P1));
  float nn = sqrtf(f3dot(n, n));
  if (nn == 0.0f) nn = 0.01f;     // nn + where(nn==0, 0.01, 0)
  return {n.x / nn, n.y / nn, n.z / nn};
}

__device__ __forceinline__ float cossim(F3 a, F3 b){
  float na = fmaxf(sqrtf(f3dot(a, a)), 1e-8f);
  float nb = fmaxf(sqrtf(f3dot(b, b)), 1e-8f);
  return f3dot(a, b) / (na * nb);
}

// order-preserving float<->key transform
__device__ __forceinline__ unsigned keyOf(float f){
  unsigned u = __float_as_uint(f);
  return u ^ ((u & 0x80000000u) ? 0xFFFFFFFFu : 0x80000000u);
}
__device__ __forceinline__ float keyInv(unsigned k){
  unsigned u = (k & 0x80000000u) ? (k ^ 0x80000000u) : ~k;
  return __uint_as_float(u);
}

// Shared block accumulation of (loss-sum, valid-count) into group gid.
__device__ __forceinline__ void blockAccum(float contrib, float validf,
                                           GroupState* st, int gid){
  __shared__ float sS;
  __shared__ float sV;
  if (threadIdx.x == 0){ sS = 0.0f; sV = 0.0f; }
  __syncthreads();
  float wS = waveSumWMMA(contrib);   // v_wmma path #1
  float wV = waveSumWMMA(validf);    // v_wmma path #2 (exact: ints <= 32 in f32)
  if ((threadIdx.x & 31) == 0){
    atomicAdd(&sS, wS);              // ds_add_f32
    atomicAdd(&sV, wV);
  }
  __syncthreads();
  if (threadIdx.x == 0){
    atomicAdd(&st[gid].S, sS);                           // global_atomic_add_f32
    atomicAdd(&st[gid].V, (unsigned)(sV + 0.5f));
  }
}

// --------------------------------- kernels ---------------------------------

__global__ void initK(GroupState* st, unsigned* hcnt, float* hsum){
  int t = blockIdx.x * 256 + threadIdx.x;
  if (t < NG * 256){ hcnt[t] = 0u; hsum[t] = 0.0f; }
  if (t < NG){
    st[t].V = 0u; st[t].S = 0.0f; st[t].prefix = 0u;
    st[t].r = 0u; st[t].done = 0u; st[t].sum_less = 0.0f;
    st[t].pad0 = 0u; st[t].pad1 = 0u;
  }
}

__global__ void planarK(const float* __restrict__ pred,
                        const float* __restrict__ km,
                        const float* __restrict__ gpl,
                        const int* __restrict__ p1,
                        const int* __restrict__ p2,
                        const int* __restrict__ p3,
                        float* __restrict__ losses, GroupState* st){
  int idx = blockIdx.x * 256 + threadIdx.x;   // grid covers exactly TOT
  int n = idx / G;                            // uniform per block (G % 256 == 0)
  float fx = km[0], fy = km[4];
  F3 P1 = unproject(pred, p1[idx], fx, fy);
  F3 P2 = unproject(pred, p2[idx], fx, fy);
  F3 P3 = unproject(pred, p3[idx], fx, fy);
  bool m = filterMask(P1, P2, P3, 0.005f);
  F3 dn = normalOf(P1, P2, P3);
  F3 gp = { gpl[n * 3 + 0], gpl[n * 3 + 1], gpl[n * 3 + 2] };
  float loss = 1.0f - fabsf(cossim(dn, gp));
  losses[idx] = m ? loss : __builtin_inff();  // invalid sorts after finite valid
  float contrib = (m && !(loss != loss)) ? loss : 0.0f;
  blockAccum(contrib, m ? 1.0f : 0.0f, st, n);
}

__global__ void nonplanarK(const float* __restrict__ pred,
                           const float* __restrict__ gt,
                           const float* __restrict__ km,
                           const int* __restrict__ q1,
                           const int* __restrict__ q2,
                           const int* __restrict__ q3,
                           float* __restrict__ losses, GroupState* st){
  int g = blockIdx.x * 256 + threadIdx.x;     // grid covers exactly GN
  float fx = km[0], fy = km[4];
  int i1 = q1[g], i2 = q2[g], i3 = q3[g];
  F3 A1 = unproject(gt, i1, fx, fy);
  F3 A2 = unproject(gt, i2, fx, fy);
  F3 A3 = unproject(gt, i3, fx, fy);
  bool m = filterMask(A1, A2, A3, 0.1f);
  F3 B1 = unproject(pred, i1, fx, fy);
  F3 B2 = unproject(pred, i2, fx, fy);
  F3 B3 = unproject(pred, i3, fx, fy);
  // replicate ref broadcast quirk: zc[c] = (point c).z==0 -> coord c of ALL points = 1e-4
  bool z0 = (B1.z == 0.0f), z1 = (B2.z == 0.0f), z2 = (B3.z == 0.0f);
  if (z0){ B1.x = 1e-4f; B2.x = 1e-4f; B3.x = 1e-4f; }
  if (z1){ B1.y = 1e-4f; B2.y = 1e-4f; B3.y = 1e-4f; }
  if (z2){ B1.z = 1e-4f; B2.z = 1e-4f; B3.z = 1e-4f; }
  float loss = 1.0f - fabsf(cossim(normalOf(B1, B2, B3), normalOf(A1, A2, A3)));
  losses[TOT + g] = m ? loss : __builtin_inff();
  float contrib = (m && !(loss != loss)) ? loss : 0.0f;
  blockAccum(contrib, m ? 1.0f : 0.0f, st, NP);
}

__global__ void prepK(GroupState* st){
  int t = threadIdx.x;
  if (t < NG){
    unsigned k = st[t].V / 4u;   // number of smallest valid losses to drop
    st[t].r = k;
    st[t].done = (k == 0u) ? 1u : 0u;
    st[t].prefix = 0u;
    st[t].sum_less = 0.0f;
  }
}

// One radix pass: per-group 256-bin histogram (count + value-sum) of the
// byte `pass` of the key, restricted to elements matching the resolved prefix.
__global__ void histK(const float* __restrict__ losses, const GroupState* __restrict__ st,
                      unsigned* hcnt, float* hsum, int pass){
  __shared__ unsigned lc[256];
  __shared__ float    ls[256];
  int t = threadIdx.x;
  lc[t] = 0u; ls[t] = 0.0f;
  __syncthreads();
  int e = blockIdx.x * 256 + t;               // < NELEM exactly
  int gid = (e < TOT) ? (e / G) : NP;         // uniform per block
  GroupState s = st[gid];
  if (!s.done){
    float lv = losses[e];
    unsigned key = keyOf(lv);
    bool match = (pass == 0) || ((key >> (32 - 8 * pass)) == s.prefix);
    if (match){
      unsigned bin = (key >> (24 - 8 * pass)) & 255u;
      atomicAdd(&lc[bin], 1u);
      atomicAdd(&ls[bin], lv);                // ds_add_f32
    }
  }
  __syncthreads();
  atomicAdd(&hcnt[gid * 256 + t], lc[t]);
  atomicAdd(&hsum[gid * 256 + t], ls[t]);
}

// Resolve one byte per group: pick smallest bin with cum >= r; accumulate
// strictly-below sums; zero histogram for next pass.
__global__ void scanK(GroupState* st, unsigned* hcnt, float* hsum){
  int g = threadIdx.x;
  if (g >= NG) return;
  GroupState s = st[g];
  if (!s.done){
    unsigned r = s.r;            // >= 1 here
    unsigned cum = 0u, b = 0u;
    float sless = 0.0f;
    for (b = 0u; b < 256u; ++b){
      unsigned c = hcnt[g * 256 + b];
      if (cum + c >= r) break;
      cum += c;
      sless += hsum[g * 256 + b];
    }
    if (b > 255u) b = 255u;      // safety clamp (unreachable by invariant)
    st[g].prefix = (s.prefix << 8) | b;
    st[g].r = r - cum;           // rank within chosen bin
    st[g].sum_less = s.sum_less + sless;
  }
  for (int i = 0; i < 256; ++i){
    hcnt[g * 256 + i] = 0u;
    hsum[g * 256 + i] = 0.0f;
  }
}

__device__ __forceinline__ float groupMean(const GroupState& s){
  unsigned k = s.V / 4u;
  float dropped = (k == 0u) ? 0.0f
                            : (s.sum_less + (float)s.r * keyInv(s.prefix));
  unsigned keep = s.V - k;
  if (keep < 1u) keep = 1u;
  return (s.S - dropped) / (float)keep;
}

__global__ void finalK(const GroupState* __restrict__ st, float* __restrict__ out){
  float planar = 0.0f;
  for (int g = 0; g < NP; ++g) planar += groupMean(st[g]);
  float res;
  if (st[NP].V > 0u) res = (planar + groupMean(st[NP])) / (float)(NP + 1);
  else               res = planar / (float)NP;
  out[0] = res;
}

// ------------------------------- launcher ----------------------------------

extern "C" void kernel_launch(void* const* d_in, const int* in_sizes, int n_in,
                              void* d_out, int out_size, void* d_ws, size_t ws_size,
                              hipStream_t stream){
  const float* pred = (const float*)d_in[0];
  const float* gt   = (const float*)d_in[1];
  const float* gpl  = (const float*)d_in[2];
  const float* km   = (const float*)d_in[3];
  const int*   p1   = (const int*)d_in[4];
  const int*   p2   = (const int*)d_in[5];
  const int*   p3   = (const int*)d_in[6];
  const int*   q1   = (const int*)d_in[7];
  const int*   q2   = (const int*)d_in[8];
  const int*   q3   = (const int*)d_in[9];
  (void)in_sizes; (void)n_in; (void)out_size; (void)ws_size;

  char* ws = (char*)d_ws;
  const size_t LOSS_BYTES = (size_t)NELEM * sizeof(float);       // 2,488,320
  const size_t ST_BYTES   = (size_t)NG * sizeof(GroupState);     // 288
  const size_t HC_BYTES   = (size_t)NG * 256 * sizeof(unsigned); // 9,216
  float*      losses = (float*)ws;
  GroupState* st     = (GroupState*)(ws + LOSS_BYTES);
  unsigned*   hcnt   = (unsigned*)(ws + LOSS_BYTES + ST_BYTES);
  float*      hsum   = (float*)(ws + LOSS_BYTES + ST_BYTES + HC_BYTES);

  initK<<<9, 256, 0, stream>>>(st, hcnt, hsum);
  planarK<<<TOT / 256, 256, 0, stream>>>(pred, km, gpl, p1, p2, p3, losses, st);
  nonplanarK<<<GN / 256, 256, 0, stream>>>(pred, gt, km, q1, q2, q3, losses, st);
  prepK<<<1, 32, 0, stream>>>(st);
  for (int pass = 0; pass < 4; ++pass){
    histK<<<NELEM / 256, 256, 0, stream>>>(losses, st, hcnt, hsum, pass);
    scanK<<<1, 32, 0, stream>>>(st, hcnt, hsum);
  }
  finalK<<<1, 1, 0, stream>>>(st, (float*)d_out);
}